// GraphBasedLSTMClassifier_70626442215574
// MI455X (gfx1250) — compile-verified
//
#include <hip/hip_runtime.h>
#include <math.h>

// ---------------- problem constants ----------------
constexpr int T_ = 16, B_ = 16, N_ = 1024, E_ = 16384;
constexpr int F_ = 128, MID_ = 16, H_ = 256;
constexpr int G_ = T_ * B_;           // 256 graphs
constexpr int KKEEP = 820;            // SAGPool keep
constexpr int KPAD  = 832;            // 52 * 16 (pad to 16-row tiles)

typedef __attribute__((ext_vector_type(2))) float v2f;
typedef __attribute__((ext_vector_type(8))) float v8f;

__device__ __forceinline__ v8f wmma4(v2f a, v2f b, v8f c) {
  // D = A(16x4,f32) * B(4x16,f32) + C(16x16,f32)
  return __builtin_amdgcn_wmma_f32_16x16x4_f32(false, a, false, b, (short)0, c,
                                               false, false);
}
__device__ __forceinline__ float sigm(float x) { return 1.0f / (1.0f + expf(-x)); }

// ---------------- init: zscat=0, deg1=1 ----------------
__global__ __launch_bounds__(256) void k_init(float* zscat, float* deg1) {
  int i = blockIdx.x * 256 + threadIdx.x;
  if (i < G_ * N_ * MID_) zscat[i] = 0.0f;
  if (i < G_ * N_)        deg1[i]  = 1.0f;
}

// ---------------- conv1: hmid = x @ W1 (WMMA f32 16x16x4) ----------------
// grid = G_*16 blocks, 128 thr (4 waves); each wave: one 16-row tile, K=128.
// x tile (contiguous 8KB) staged via CDNA5 async Global->LDS engine (ASYNCcnt).
__global__ __launch_bounds__(128) void k_conv1(const float* __restrict__ x,
                                               const float* __restrict__ W1,
                                               float* __restrict__ hmid) {
  __shared__ float sW[F_][MID_];        // 8 KB
  __shared__ float sX[4][16][F_];       // 32 KB
  const int g = blockIdx.x >> 4;
  const int rb_blk = (blockIdx.x & 15) * 64;
  const int tid = threadIdx.x, wave = tid >> 5, lane = tid & 31;
  for (int i = tid; i < F_ * MID_; i += 128) sW[i / MID_][i % MID_] = W1[i];
  const int rowbase = rb_blk + wave * 16;
  const float4* xt4 = (const float4*)(x + ((size_t)g * N_ + rowbase) * F_);
  // low 32 bits of a shared-aperture flat pointer == LDS byte offset
  const unsigned ldsbase = (unsigned)(uintptr_t)&sX[wave][0][0];
  for (int i = lane; i < 16 * F_ / 4; i += 32) {
    const float4* gp = xt4 + i;             // per-lane global address (GV mode)
    unsigned la = ldsbase + i * 16;         // per-lane LDS dest address
    asm volatile("global_load_async_to_lds_b128 %0, %1, off"
                 :: "v"(la), "v"(gp) : "memory");
  }
  asm volatile("s_wait_asynccnt 0" ::: "memory");
  __syncthreads();
  const int half = lane >> 4, lr = lane & 15;
  v8f acc = {0.f,0.f,0.f,0.f,0.f,0.f,0.f,0.f};
  for (int kk = 0; kk < F_; kk += 4) {
    const int k0 = kk + half * 2;
    v2f a, b;
    a.x = sX[wave][lr][k0];  a.y = sX[wave][lr][k0 + 1];
    b.x = sW[k0][lr];        b.y = sW[k0 + 1][lr];
    acc = wmma4(a, b, acc);
  }
  float* out = hmid + ((size_t)g * N_ + rowbase) * MID_;
  for (int r = 0; r < 8; ++r) out[(r + half * 8) * MID_ + lr] = acc[r];
}

// ---------------- deg1: in-degree + 1 ----------------
__global__ __launch_bounds__(256) void k_deg1(const int* __restrict__ ei, float* deg1) {
  int idx = blockIdx.x * 256 + threadIdx.x;
  if (idx >= G_ * E_) return;
  int g = idx / E_, e = idx - g * E_;
  int dst = ei[(size_t)g * 2 * E_ + E_ + e];
  atomicAdd(&deg1[(size_t)g * N_ + dst], 1.0f);
}

// ---------------- scatter1: zscat[dst] += hmid[src] * coef ----------------
__global__ __launch_bounds__(256) void k_scatter1(const int* __restrict__ ei,
                                                  const float* __restrict__ hmid,
                                                  const float* __restrict__ deg1,
                                                  float* zscat) {
  int idx = blockIdx.x * 256 + threadIdx.x;
  if (idx >= G_ * E_) return;
  int g = idx / E_, e = idx - g * E_;
  const int* eg = ei + (size_t)g * 2 * E_;
  int src = eg[e], dst = eg[E_ + e];
  float coef = rsqrtf(deg1[(size_t)g * N_ + src]) * rsqrtf(deg1[(size_t)g * N_ + dst]);
  const float4* hs = (const float4*)(hmid + ((size_t)g * N_ + src) * MID_);
  float* zd = zscat + ((size_t)g * N_ + dst) * MID_;
  for (int q = 0; q < 4; ++q) {
    float4 v = hs[q];
    atomicAdd(zd + 4 * q + 0, v.x * coef);
    atomicAdd(zd + 4 * q + 1, v.y * coef);
    atomicAdd(zd + 4 * q + 2, v.z * coef);
    atomicAdd(zd + 4 * q + 3, v.w * coef);
  }
}

// ---------------- combine1: h1 = relu(zscat + hmid/deg + b); score init ----------
__global__ __launch_bounds__(256) void k_combine1(const float* __restrict__ hmid,
                                                  float* zscat /* in scat / out h1 */,
                                                  const float* __restrict__ deg1,
                                                  const float* __restrict__ b1,
                                                  const float* __restrict__ Wr,
                                                  const float* __restrict__ sagb,
                                                  float* score) {
  int n = blockIdx.x * 256 + threadIdx.x;
  if (n >= G_ * N_) return;
  float dinv2 = 1.0f / deg1[n];
  const float* hm = hmid + (size_t)n * MID_;
  float* zz = zscat + (size_t)n * MID_;
  float sc = sagb[0];
  for (int c = 0; c < MID_; ++c) {
    float v = zz[c] + hm[c] * dinv2 + b1[c];
    v = v > 0.f ? v : 0.f;
    zz[c] = v;                 // overwrite scatter buffer with h1
    sc += v * Wr[c];           // self part of SAG score
  }
  score[n] = sc;
}

// ---------------- score scatter: score[dst] += h1[src].Wl --------------------
__global__ __launch_bounds__(256) void k_scoreScat(const int* __restrict__ ei,
                                                   const float* __restrict__ h1,
                                                   const float* __restrict__ Wl,
                                                   float* score) {
  int idx = blockIdx.x * 256 + threadIdx.x;
  if (idx >= G_ * E_) return;
  int g = idx / E_, e = idx - g * E_;
  const int* eg = ei + (size_t)g * 2 * E_;
  int src = eg[e], dst = eg[E_ + e];
  const float* hr = h1 + ((size_t)g * N_ + src) * MID_;
  float s = 0.f;
  for (int c = 0; c < MID_; ++c) s += hr[c] * Wl[c];
  atomicAdd(&score[(size_t)g * N_ + dst], s);
}

// ---------------- top-k via bitonic sort; build xp, newidx, deg2 init --------
__global__ __launch_bounds__(512) void k_topk(const float* __restrict__ score,
                                              const float* __restrict__ h1,
                                              float* xp, int* newidx, float* deg2) {
  __shared__ float ss[N_];
  __shared__ int   si[N_];
  const int g = blockIdx.x, tid = threadIdx.x;
  for (int i = tid; i < N_; i += 512) { ss[i] = score[(size_t)g * N_ + i]; si[i] = i; }
  __syncthreads();
  for (int k = 2; k <= N_; k <<= 1) {
    for (int j = k >> 1; j > 0; j >>= 1) {
      int i = ((tid & ~(j - 1)) << 1) | (tid & (j - 1));
      int p = i | j;
      float a = ss[i], b = ss[p];
      int ia = si[i], ib = si[p];
      bool aFirst = (a > b) || (a == b && ia < ib);   // rank order (desc, tie: low idx)
      bool desc = ((i & k) == 0);
      if (desc ? !aFirst : aFirst) { ss[i] = b; ss[p] = a; si[i] = ib; si[p] = ia; }
      __syncthreads();
    }
  }
  for (int n = tid; n < N_; n += 512) newidx[(size_t)g * N_ + n] = -1;
  __syncthreads();
  for (int r = tid; r < KKEEP; r += 512) newidx[(size_t)g * N_ + si[r]] = r;
  for (int r = tid; r < KPAD; r += 512) {
    float* xr = xp + ((size_t)g * KPAD + r) * MID_;
    deg2[(size_t)g * KPAD + r] = 1.0f;
    if (r < KKEEP) {
      float gate = tanhf(ss[r]);
      const float* hr = h1 + ((size_t)g * N_ + si[r]) * MID_;
      for (int c = 0; c < MID_; ++c) xr[c] = hr[c] * gate;
    } else {
      for (int c = 0; c < MID_; ++c) xr[c] = 0.f;
    }
  }
}

// ---------------- deg2 over remapped valid edges ----------------
__global__ __launch_bounds__(256) void k_deg2(const int* __restrict__ ei,
                                              const int* __restrict__ newidx, float* deg2) {
  int idx = blockIdx.x * 256 + threadIdx.x;
  if (idx >= G_ * E_) return;
  int g = idx / E_, e = idx - g * E_;
  const int* eg = ei + (size_t)g * 2 * E_;
  int ns = newidx[(size_t)g * N_ + eg[e]];
  int nd = newidx[(size_t)g * N_ + eg[E_ + e]];
  if (ns >= 0 && nd >= 0) atomicAdd(&deg2[(size_t)g * KPAD + nd], 1.0f);
}

// ---------------- zs2 = xp/deg2 (self term), pad rows zero ----------------
__global__ __launch_bounds__(256) void k_zs2init(const float* __restrict__ xp,
                                                 const float* __restrict__ deg2, float* zs2) {
  int idx = blockIdx.x * 256 + threadIdx.x;
  if (idx >= G_ * KPAD) return;
  int r = idx % KPAD;
  float* z = zs2 + (size_t)idx * MID_;
  if (r < KKEEP) {
    float f = 1.0f / deg2[idx];
    const float* xr = xp + (size_t)idx * MID_;
    for (int c = 0; c < MID_; ++c) z[c] = xr[c] * f;
  } else {
    for (int c = 0; c < MID_; ++c) z[c] = 0.f;
  }
}

// ---------------- scatter2 in 16-dim space (GEMM commutes) ----------------
__global__ __launch_bounds__(256) void k_scatter2(const int* __restrict__ ei,
                                                  const int* __restrict__ newidx,
                                                  const float* __restrict__ xp,
                                                  const float* __restrict__ deg2, float* zs2) {
  int idx = blockIdx.x * 256 + threadIdx.x;
  if (idx >= G_ * E_) return;
  int g = idx / E_, e = idx - g * E_;
  const int* eg = ei + (size_t)g * 2 * E_;
  int ns = newidx[(size_t)g * N_ + eg[e]];
  int nd = newidx[(size_t)g * N_ + eg[E_ + e]];
  if (ns < 0 || nd < 0) return;
  float coef = rsqrtf(deg2[(size_t)g * KPAD + ns]) * rsqrtf(deg2[(size_t)g * KPAD + nd]);
  const float4* xs = (const float4*)(xp + ((size_t)g * KPAD + ns) * MID_);
  float* zd = zs2 + ((size_t)g * KPAD + nd) * MID_;
  for (int q = 0; q < 4; ++q) {
    float4 v = xs[q];
    atomicAdd(zd + 4 * q + 0, v.x * coef);
    atomicAdd(zd + 4 * q + 1, v.y * coef);
    atomicAdd(zd + 4 * q + 2, v.z * coef);
    atomicAdd(zd + 4 * q + 3, v.w * coef);
  }
}

// ---------------- conv2 fused: emb = mean(relu(zs2 @ W2 + b2)) ----------------
// grid = G_*2, 256 thr (8 waves); each wave owns 16 output columns, loops 52 M-tiles
__global__ __launch_bounds__(256) void k_conv2pool(const float* __restrict__ zs2,
                                                   const float* __restrict__ W2,
                                                   const float* __restrict__ b2,
                                                   float* __restrict__ emb) {
  const int g = blockIdx.x >> 1;
  const int wave = threadIdx.x >> 5, lane = threadIdx.x & 31;
  const int half = lane >> 4, lr = lane & 15;
  const int colbase = ((blockIdx.x & 1) * 8 + wave) * 16;
  const float* z = zs2 + (size_t)g * KPAD * MID_;
  v2f bf[4];
  for (int s = 0; s < 4; ++s) {
    int k0 = s * 4 + half * 2;
    bf[s].x = W2[k0 * H_ + colbase + lr];
    bf[s].y = W2[(k0 + 1) * H_ + colbase + lr];
  }
  const float bias = b2[colbase + lr];
  float colsum = 0.f;
  for (int mt = 0; mt < KPAD / 16; ++mt) {
    const float* zr = z + (size_t)mt * 16 * MID_;
    v8f acc = {0.f,0.f,0.f,0.f,0.f,0.f,0.f,0.f};
    for (int s = 0; s < 4; ++s) {
      int k0 = s * 4 + half * 2;
      v2f a; a.x = zr[lr * MID_ + k0]; a.y = zr[lr * MID_ + k0 + 1];
      acc = wmma4(a, bf[s], acc);
    }
    for (int r = 0; r < 8; ++r) {
      int m = mt * 16 + r + half * 8;
      if (m < KKEEP) { float v = acc[r] + bias; colsum += (v > 0.f ? v : 0.f); }
    }
  }
  colsum += __shfl_xor(colsum, 16, 32);          // combine M-halves (same column)
  if (half == 0) emb[(size_t)g * H_ + colbase + lr] = colsum * (1.0f / KKEEP);
}

// ---------------- LSTM + classifier: 1 block, 16 waves, c resident in VGPRs ----
__global__ __launch_bounds__(512) void k_lstm(const float* __restrict__ emb,
                                              const float* __restrict__ Wih,
                                              const float* __restrict__ Whh,
                                              const float* __restrict__ bih,
                                              const float* __restrict__ bhh,
                                              const float* __restrict__ clsW,
                                              const float* __restrict__ clsb,
                                              float* __restrict__ out) {
  __shared__ float hbuf[B_][H_];                 // 16 KB
  const int tid = threadIdx.x, wave = tid >> 5, lane = tid & 31;
  const int half = lane >> 4, lr = lane & 15;
  const int jbase = wave * 16;                   // this wave's hidden columns
  for (int i = tid; i < B_ * H_; i += 512) ((float*)hbuf)[i] = 0.f;
  v8f cfrag = {0.f,0.f,0.f,0.f,0.f,0.f,0.f,0.f};
  __syncthreads();
  for (int t = 0; t < T_; ++t) {
    const float* xt = emb + (size_t)t * B_ * H_;
    v8f gfr[4];
    for (int gt = 0; gt < 4; ++gt) {
      const int gcol = gt * H_ + jbase + lr;     // row of W_ih/W_hh
      const float bsum = bih[gcol] + bhh[gcol];
      v8f acc = {bsum, bsum, bsum, bsum, bsum, bsum, bsum, bsum};
      const float* wi = Wih + (size_t)gcol * H_;
      const float* wh = Whh + (size_t)gcol * H_;
      for (int kk = 0; kk < H_; kk += 4) {       // input contribution
        int k0 = kk + half * 2;
        v2f a, b;
        a.x = xt[lr * H_ + k0]; a.y = xt[lr * H_ + k0 + 1];
        b.x = wi[k0];           b.y = wi[k0 + 1];
        acc = wmma4(a, b, acc);
      }
      for (int kk = 0; kk < H_; kk += 4) {       // recurrent contribution
        int k0 = kk + half * 2;
        v2f a, b;
        a.x = hbuf[lr][k0]; a.y = hbuf[lr][k0 + 1];
        b.x = wh[k0];       b.y = wh[k0 + 1];
        acc = wmma4(a, b, acc);
      }
      gfr[gt] = acc;
    }
    __syncthreads();                              // all hbuf reads done
    v8f hnew;
    for (int r = 0; r < 8; ++r) {
      float iv = sigm(gfr[0][r]);
      float fv = sigm(gfr[1][r]);
      float gv = tanhf(gfr[2][r]);
      float ov = sigm(gfr[3][r]);
      float c = fv * cfrag[r] + iv * gv;
      cfrag[r] = c;
      hnew[r] = ov * tanhf(c);
    }
    for (int r = 0; r < 8; ++r) hbuf[r + half * 8][jbase + lr] = hnew[r];
    __syncthreads();
  }
  if (tid < B_) {                                 // classifier: out = hT @ clsW + b
    float s = clsb[0];
    for (int k = 0; k < H_; ++k) s += hbuf[tid][k] * clsW[k];
    out[tid] = s;
  }
}

// ============================ launch ============================
extern "C" void kernel_launch(void* const* d_in, const int* in_sizes, int n_in,
                              void* d_out, int out_size, void* d_ws, size_t ws_size,
                              hipStream_t stream) {
  (void)in_sizes; (void)n_in; (void)out_size; (void)ws_size;
  const float* x      = (const float*)d_in[0];
  const int*   ei     = (const int*)  d_in[1];
  const float* c1W    = (const float*)d_in[2];
  const float* c1b    = (const float*)d_in[3];
  const float* sagWl  = (const float*)d_in[4];
  const float* sagWr  = (const float*)d_in[5];
  const float* sagb   = (const float*)d_in[6];
  const float* c2W    = (const float*)d_in[7];
  const float* c2b    = (const float*)d_in[8];
  const float* Wih    = (const float*)d_in[9];
  const float* Whh    = (const float*)d_in[10];
  const float* bih    = (const float*)d_in[11];
  const float* bhh    = (const float*)d_in[12];
  const float* clsW   = (const float*)d_in[13];
  const float* clsb   = (const float*)d_in[14];
  float* outp = (float*)d_out;

  const size_t GN = (size_t)G_ * N_;
  float* ws    = (float*)d_ws;
  float* hmid  = ws;                                  // GN*16
  float* h1    = hmid + GN * MID_;                    // GN*16 (scatter acc -> h1)
  float* deg1  = h1 + GN * MID_;                      // GN
  float* score = deg1 + GN;                           // GN
  float* xp    = score + GN;                          // G*KPAD*16
  float* deg2  = xp + (size_t)G_ * KPAD * MID_;       // G*KPAD
  float* zs2   = deg2 + (size_t)G_ * KPAD;            // G*KPAD*16
  float* emb   = zs2 + (size_t)G_ * KPAD * MID_;      // G*H
  int*  newidx = (int*)(emb + (size_t)G_ * H_);       // GN ints

  auto cdiv = [](int a, int b) { return (a + b - 1) / b; };
  const int GE = G_ * E_;

  k_init     <<<cdiv(G_ * N_ * MID_, 256), 256, 0, stream>>>(h1, deg1);
  k_conv1    <<<G_ * 16, 128, 0, stream>>>(x, c1W, hmid);
  k_deg1     <<<cdiv(GE, 256), 256, 0, stream>>>(ei, deg1);
  k_scatter1 <<<cdiv(GE, 256), 256, 0, stream>>>(ei, hmid, deg1, h1);
  k_combine1 <<<cdiv(G_ * N_, 256), 256, 0, stream>>>(hmid, h1, deg1, c1b, sagWr, sagb, score);
  k_scoreScat<<<cdiv(GE, 256), 256, 0, stream>>>(ei, h1, sagWl, score);
  k_topk     <<<G_, 512, 0, stream>>>(score, h1, xp, newidx, deg2);
  k_deg2     <<<cdiv(GE, 256), 256, 0, stream>>>(ei, newidx, deg2);
  k_zs2init  <<<cdiv(G_ * KPAD, 256), 256, 0, stream>>>(xp, deg2, zs2);
  k_scatter2 <<<cdiv(GE, 256), 256, 0, stream>>>(ei, newidx, xp, deg2, zs2);
  k_conv2pool<<<G_ * 2, 256, 0, stream>>>(zs2, c2W, c2b, emb);
  k_lstm     <<<1, 512, 0, stream>>>(emb, Wih, Whh, bih, bhh, clsW, clsb, outp);
}